// Integrator_41850161332390
// MI455X (gfx1250) — compile-verified
//
#include <hip/hip_runtime.h>
#include <math.h>

// Voxel-pooling integrator for MI455X (gfx1250).
// Bandwidth-bound scatter/gather: no matrix contraction -> WMMA N/A.
// CDNA5 levers: fp32 global atomics (pool scatter-add resolved at the L2
// atomic units; both 64MB pooling caches fit in the 192MB L2), NT temporal
// hints on all single-use streams so they don't evict the caches, wave32
// 256-thread blocks.
//
// NOTE: k_scatter_pool is deliberately FIRST in this file so the disasm
// snippet shows the atomic lowering (want: global_atomic_add_f32, not a
// cmpswap loop) and the NT load hints.

typedef __attribute__((ext_vector_type(4))) float f4;

// ---------------------------------------------------------------------------
// Scatter-add pooling. Per point: bounds-check, flatten index, atomically
// accumulate feature value and weight 1.0 into the L2-resident caches.
// Streaming inputs (feature, indices) read non-temporally.
// ---------------------------------------------------------------------------
__global__ void k_scatter_pool(const float* __restrict__ feat,
                               const int*   __restrict__ idx,
                               float* __restrict__ fcache,
                               float* __restrict__ wcache,
                               int P, int X, int Y, int Z) {
    int p = blockIdx.x * blockDim.x + threadIdx.x;
    if (p >= P) return;

    // 12B/lane; wave reads 384B contiguous -> coalesced.
    int x = __builtin_nontemporal_load(idx + 3 * (long)p + 0);
    int y = __builtin_nontemporal_load(idx + 3 * (long)p + 1);
    int z = __builtin_nontemporal_load(idx + 3 * (long)p + 2);

    bool valid = ((unsigned)x < (unsigned)X) &
                 ((unsigned)y < (unsigned)Y) &
                 ((unsigned)z < (unsigned)Z);
    if (!valid) return;  // invalid points only touched the trimmed dummy slot

    long flat = ((long)x * Y + (long)y) * Z + (long)z;
    float f = __builtin_nontemporal_load(feat + p);

    // Want: global_atomic_add_f32 (non-returning, RMW at L2 atomic units).
    unsafeAtomicAdd(fcache + flat, f);
    unsafeAtomicAdd(wcache + flat, 1.0f);
}

// ---------------------------------------------------------------------------
// Zero the pooling caches in workspace (2*V floats).
// Regular (RT) stores so the lines land dirty in L2, ready for L2 atomics.
// ---------------------------------------------------------------------------
__global__ void k_zero_ws(f4* __restrict__ ws, int n4) {
    int t = blockIdx.x * blockDim.x + threadIdx.x;
    if (t < n4) {
        f4 z = {0.f, 0.f, 0.f, 0.f};
        ws[t] = z;
    }
}

// ---------------------------------------------------------------------------
// Voxel-parallel finalize. Equivalent to the reference's per-point gather +
// scatter-set: a voxel is updated iff some valid point mapped to it
// (wcache[v] > 0), and every such point writes the identical value
//   fv*cv + (fcache/wcache)/(cv+1).
// Fully coalesced float4 traffic; fv/cv/out are NT (single-use streams),
// caches are RT (should hit in L2).
// ---------------------------------------------------------------------------
__global__ void k_finalize(const f4* __restrict__ fcache,
                           const f4* __restrict__ wcache,
                           const f4* __restrict__ fv,
                           const f4* __restrict__ cv,
                           f4* __restrict__ out_fv,
                           f4* __restrict__ out_cv,
                           int n4) {
    int t = blockIdx.x * blockDim.x + threadIdx.x;
    if (t >= n4) return;

    f4 s = fcache[t];                          // pooled sums   (L2 resident)
    f4 w = wcache[t];                          // pooled counts (L2 resident)
    f4 f = __builtin_nontemporal_load(fv + t); // original volume (stream)
    f4 c = __builtin_nontemporal_load(cv + t); // original counts (stream)

    f4 of, oc;
#pragma unroll
    for (int i = 0; i < 4; ++i) {
        float wi = w[i], fi = f[i], ci = c[i];
        if (wi > 0.f) {
            float cu   = ci + 1.0f;
            float pool = s[i] / wi;            // precise IEEE div (matches ref)
            of[i] = fi * ci + pool / cu;
            oc[i] = cu;
        } else {
            of[i] = fi;
            oc[i] = ci;
        }
    }
    __builtin_nontemporal_store(of, out_fv + t);
    __builtin_nontemporal_store(oc, out_cv + t);
}

// ---------------------------------------------------------------------------
// Host launcher. Inputs (setup_inputs order):
//   d_in[0] feature        float32 [N,H,W]   (P elements)
//   d_in[1] indices        int32   [N,H,W,3] (3P elements)
//   d_in[2] feature_volume float32 [X,Y,Z]   (V elements)
//   d_in[3] count_volume   float32 [X,Y,Z]   (V elements)
// Output: fv_new (V floats) ++ cv_new (V floats), concatenated flat.
// Workspace: fcache (V floats) ++ wcache (V floats) = 2*V*4 bytes.
// ---------------------------------------------------------------------------
extern "C" void kernel_launch(void* const* d_in, const int* in_sizes, int n_in,
                              void* d_out, int out_size, void* d_ws, size_t ws_size,
                              hipStream_t stream) {
    const float* feat = (const float*)d_in[0];
    const int*   idx  = (const int*)  d_in[1];
    const float* fv   = (const float*)d_in[2];
    const float* cv   = (const float*)d_in[3];

    const long P = in_sizes[0];
    const long V = in_sizes[2];
    const int  X = (int)llrint(cbrt((double)V));  // cube volume (256^3)

    float* fcache = (float*)d_ws;
    float* wcache = fcache + V;

    const int BLK = 256;  // 8 wave32 waves per block

    // 1) zero the pooling caches (2*V floats, V divisible by 4)
    int n4z = (int)((2 * V) / 4);
    k_zero_ws<<<(n4z + BLK - 1) / BLK, BLK, 0, stream>>>((f4*)d_ws, n4z);

    // 2) scatter-add pooling over all points
    k_scatter_pool<<<(int)((P + BLK - 1) / BLK), BLK, 0, stream>>>(
        feat, idx, fcache, wcache, (int)P, X, X, X);

    // 3) voxel-parallel finalize into concatenated output
    int n4 = (int)(V / 4);
    k_finalize<<<(n4 + BLK - 1) / BLK, BLK, 0, stream>>>(
        (const f4*)fcache, (const f4*)wcache,
        (const f4*)fv, (const f4*)cv,
        (f4*)d_out, (f4*)d_out + n4, n4);
}